// SRSChannelEstimator_16509854286590
// MI455X (gfx1250) — compile-verified
//
#include <hip/hip_runtime.h>
#include <hip/hip_bf16.h>
#include <math.h>

#define LSEQ   393216
#define L4     (LSEQ / 4)
#define NPORT  32
#define NQ     12
#define NWIN   21
#define TILE   1536
#define HALO   4
#define TILE_LS (TILE + 2 * HALO)   // 1544
#define NAVG   386                  // avg groups covering tile + interp halo
#define TWO_PI 6.283185307179586f
#define TWO_PI_OVER_L (TWO_PI / (float)LSEQ)

// workspace layout (in 4-byte units)
#define WSF_NOISE 0
#define WSF_WIN   8            // 12*21*2 = 504 floats -> [8, 512)
#define WSI_M     512          // 32 int32
#define WSI_T     544          // 32 int32
#define WS_ZERO_N 512

typedef __attribute__((ext_vector_type(2))) float v2f;
typedef __attribute__((ext_vector_type(8))) float v8f;

#define AS1 __attribute__((address_space(1)))
#define AS3 __attribute__((address_space(3)))

#if defined(__has_builtin)
#if __has_builtin(__builtin_amdgcn_global_load_async_to_lds_b32)
#define HAVE_ASYNC_LDS 1
#endif
#endif

// exact (m*n) mod L with L = 3 * 2^17:  x mod L = ((x>>17) mod 3)<<17 | (x & 0x1FFFF)
// (x < L^2 < 2^38, so x>>17 fits in u32; only a u32 %3 remains -> mulhi sequence)
__device__ __forceinline__ unsigned mod_L(unsigned long long x) {
    const unsigned low = (unsigned)x & 0x1FFFFu;
    const unsigned hi  = (unsigned)(x >> 17);
    return ((hi % 3u) << 17) | low;
}

// exact phase: 2*pi*((m*n) mod L)/L, all operands non-negative, m,n < L
__device__ __forceinline__ float phase_of(unsigned m, unsigned n) {
    const unsigned r = mod_L((unsigned long long)m * (unsigned long long)n);
    return (float)r * TWO_PI_OVER_L;
}

// ---------------------------------------------------------------- init
__global__ void k_init_ws(float* ws) {
    for (int i = threadIdx.x; i < WS_ZERO_N; i += blockDim.x) ws[i] = 0.0f;
}

// ---------------------------------------------------------------- noise power partial sum
__global__ void k_noise(const float* __restrict__ lsr, const float* __restrict__ lsi,
                        float* __restrict__ ws) {
    __shared__ float red[256];
    float acc = 0.0f;
    for (int i = blockIdx.x * blockDim.x + threadIdx.x; i < LSEQ - 1;
         i += gridDim.x * blockDim.x) {
        const float dr = lsr[i + 1] - lsr[i];
        const float di = lsi[i + 1] - lsi[i];
        acc += dr * dr + di * di;
    }
    red[threadIdx.x] = acc;
    __syncthreads();
    for (int off = 128; off > 0; off >>= 1) {
        if (threadIdx.x < off) red[threadIdx.x] += red[threadIdx.x + off];
        __syncthreads();
    }
    if (threadIdx.x == 0) atomicAdd(&ws[WSF_NOISE], red[0]);
}

// ---------------------------------------------------------------- direct DFT at the 12*21 search points
// h[t] ~ sum_n ls[n] * e^{+i 2pi n t / L},  t = q*L/12 + d, d in [-10,10]
__global__ void k_win_dft(const float* __restrict__ lsr, const float* __restrict__ lsi,
                          float* __restrict__ ws) {
    const int q = blockIdx.y;
    __shared__ float w12c[12], w12s[12];
    __shared__ float redr[256], reds[256];
    if (threadIdx.x < 12) {
        float s, c;
        __sincosf((float)threadIdx.x * (TWO_PI / 12.0f), &s, &c);
        w12s[threadIdx.x] = s;
        w12c[threadIdx.x] = c;
    }
    __syncthreads();

    float accr[NWIN], acci[NWIN];
#pragma unroll
    for (int d = 0; d < NWIN; ++d) { accr[d] = 0.0f; acci[d] = 0.0f; }

    const int n_base = blockIdx.x * 4096;
    for (int it = 0; it < 16; ++it) {
        const int n = n_base + it * 256 + threadIdx.x;
        const float xr = lsr[n];
        const float xi = lsi[n];
        const int r12 = ((n % 12) * q) % 12;
        const float wc = w12c[r12], wsn = w12s[r12];
        const float wr = xr * wc - xi * wsn;       // ls[n] * w12^{nq}
        const float wi = xr * wsn + xi * wc;
        const float th = (float)n * TWO_PI_OVER_L; // base angle, in [0, 2pi)
        float ts, tc;  __sincosf(th, &ts, &tc);          // rotation e^{+i th}
        float psn, pc; __sincosf(-10.0f * th, &psn, &pc); // start at d = -10
#pragma unroll
        for (int d = 0; d < NWIN; ++d) {
            accr[d] += wr * pc - wi * psn;
            acci[d] += wr * psn + wi * pc;
            const float npc = pc * tc - psn * ts;
            psn = pc * ts + psn * tc;
            pc = npc;
        }
    }

#pragma unroll
    for (int d = 0; d < NWIN; ++d) {
        redr[threadIdx.x] = accr[d];
        reds[threadIdx.x] = acci[d];
        __syncthreads();
        for (int off = 128; off > 0; off >>= 1) {
            if (threadIdx.x < off) {
                redr[threadIdx.x] += redr[threadIdx.x + off];
                reds[threadIdx.x] += reds[threadIdx.x + off];
            }
            __syncthreads();
        }
        if (threadIdx.x == 0) {
            atomicAdd(&ws[WSF_WIN + (q * NWIN + d) * 2 + 0], redr[0]);
            atomicAdd(&ws[WSF_WIN + (q * NWIN + d) * 2 + 1], reds[0]);
        }
        __syncthreads();
    }
}

// ---------------------------------------------------------------- per-port timing-offset selection
__global__ void k_select(const int* __restrict__ shifts, float* __restrict__ ws) {
    const int u = threadIdx.x;
    if (u >= NPORT) return;
    int* wsi = (int*)ws;
    const int s = shifts[u];
    const int q = (12 - (s % 12)) % 12;
    float best = -1.0f;
    int bi = 0;
    for (int i = 0; i < NWIN; ++i) {
        const float re = ws[WSF_WIN + (q * NWIN + i) * 2 + 0];
        const float im = ws[WSF_WIN + (q * NWIN + i) * 2 + 1];
        const float mag = re * re + im * im;
        if (mag > best) { best = mag; bi = i; }   // strict >: first max, matches argmax
    }
    const int T = bi - 10;
    const long long m = (long long)T + (long long)q * (LSEQ / 12);
    wsi[WSI_M + u] = (int)(((m % LSEQ) + LSEQ) % LSEQ);
    wsi[WSI_T + u] = (int)(((T % LSEQ) + LSEQ) % LSEQ);
}

// ---------------------------------------------------------------- helpers for the fused kernel
__device__ __forceinline__ void port_avg(const float* ls_re, const float* ls_im,
                                         float* avg_re, float* avg_im,
                                         int base, unsigned m_mod, float stc, float sts) {
    const int base4 = base >> 2;
    for (int a = threadIdx.x; a < NAVG; a += 256) {
        int j = base4 - 1 + a;
        j = j < 0 ? 0 : (j > (L4 - 1) ? (L4 - 1) : j);   // edge clamp == jnp.interp clamp
        const int n0 = 4 * j;
        const int l0 = n0 - base + HALO;
        float s, c; __sincosf(phase_of(m_mod, (unsigned)n0), &s, &c);
        float ar = 0.0f, ai = 0.0f;
#pragma unroll
        for (int k = 0; k < 4; ++k) {
            const float xr = ls_re[l0 + k], xi = ls_im[l0 + k];
            ar += xr * c - xi * s;                // ls * ph_m
            ai += xr * s + xi * c;
            const float nc = c * stc - s * sts;   // rotate by e^{i 2pi m / L}
            s = c * sts + s * stc;
            c = nc;
        }
        avg_re[a] = ar * 0.25f;
        avg_im[a] = ai * 0.25f;
    }
}

__device__ __forceinline__ void interp_at(const float* avg_re, const float* avg_im,
                                          int n, int base4, float* hr, float* hi) {
    const int j = (n >= 2) ? ((n - 2) >> 2) : -1;        // floor((n-1.5)/4)
    const float w = ((float)n - 1.5f - 4.0f * (float)j) * 0.25f;
    const int a = j - base4 + 1;                         // local avg index
    const float r0 = avg_re[a], i0 = avg_im[a];
    const float r1 = avg_re[a + 1], i1 = avg_im[a + 1];
    *hr = r0 + (r1 - r0) * w;
    *hi = i0 + (i1 - i0) * w;
}

// ---------------------------------------------------------------- fused main kernel
// per 1536-sample tile: de-shift+avg+interp for 32 ports, residual, h_wr,
// block-MMSE via V_WMMA_F32_16X16X4_F32, align by conj(ph_T), store.
__global__ void __launch_bounds__(256)
k_main(const float* __restrict__ lsr, const float* __restrict__ lsi,
       const float* __restrict__ ws, float* __restrict__ out) {
    __shared__ float ls_re[TILE_LS], ls_im[TILE_LS];
    __shared__ float avg_re[NAVG + 2], avg_im[NAVG + 2];
    __shared__ float rec_re[TILE], rec_im[TILE];    // recon accum, then residual
    __shared__ float wr_re[TILE], wr_im[TILE];

    const int* wsi = (const int*)ws;
    const int base = blockIdx.x * TILE;
    const int base4 = base >> 2;

    // warm L2 for a neighboring tile (global_prefetch_b8)
    if (blockIdx.x + 1 < gridDim.x && threadIdx.x < 8) {
        __builtin_prefetch(lsr + base + TILE + threadIdx.x * 64, 0, 1);
        __builtin_prefetch(lsi + base + TILE + threadIdx.x * 64, 0, 1);
    }

    // stage 0: ls tile (+halo, clamped) into LDS via async-to-LDS when available;
    // the recon-accumulator zeroing overlaps the in-flight async copies.
#ifdef HAVE_ASYNC_LDS
    for (int i = threadIdx.x; i < TILE_LS; i += 256) {
        int g = base - HALO + i;
        g = g < 0 ? 0 : (g > LSEQ - 1 ? LSEQ - 1 : g);
        __builtin_amdgcn_global_load_async_to_lds_b32(
            (AS1 int*)(size_t)(lsr + g),
            (AS3 int*)(unsigned)(size_t)&ls_re[i], 0, 0);
        __builtin_amdgcn_global_load_async_to_lds_b32(
            (AS1 int*)(size_t)(lsi + g),
            (AS3 int*)(unsigned)(size_t)&ls_im[i], 0, 0);
    }
    for (int i = threadIdx.x; i < TILE; i += 256) { rec_re[i] = 0.0f; rec_im[i] = 0.0f; }
#if defined(__has_builtin) && __has_builtin(__builtin_amdgcn_s_wait_asynccnt)
    __builtin_amdgcn_s_wait_asynccnt(0);
#else
    asm volatile("s_wait_asynccnt 0" ::: "memory");
#endif
#else
    for (int i = threadIdx.x; i < TILE_LS; i += 256) {
        int g = base - HALO + i;
        g = g < 0 ? 0 : (g > LSEQ - 1 ? LSEQ - 1 : g);
        ls_re[i] = lsr[g];
        ls_im[i] = lsi[g];
    }
    for (int i = threadIdx.x; i < TILE; i += 256) { rec_re[i] = 0.0f; rec_im[i] = 0.0f; }
#endif
    __syncthreads();

    const float nsum = ws[WSF_NOISE];
    const float sigma2 = nsum / (float)(LSEQ - 1) * 0.5f;
    const float coef = 1.0f / (1.0f + sigma2);          // M = C (C+R)^-1 = coef * I

    // ---- pass 1: accumulate h_recon = sum_u h_interp * conj(ph_m) ----
    for (int u = 0; u < NPORT; ++u) {
        const unsigned m_mod = (unsigned)wsi[WSI_M + u];
        float sts, stc; __sincosf((float)m_mod * TWO_PI_OVER_L, &sts, &stc);

        port_avg(ls_re, ls_im, avg_re, avg_im, base, m_mod, stc, sts);
        __syncthreads();

        const int nl0 = threadIdx.x * 6;
        float s, c; __sincosf(phase_of(m_mod, (unsigned)(base + nl0)), &s, &c);
#pragma unroll
        for (int e = 0; e < 6; ++e) {
            const int nl = nl0 + e;
            float hr, hi;
            interp_at(avg_re, avg_im, base + nl, base4, &hr, &hi);
            rec_re[nl] += hr * c + hi * s;              // h_interp * conj(ph_m)
            rec_im[nl] += hi * c - hr * s;
            const float nc = c * stc - s * sts;
            s = c * sts + s * stc;
            c = nc;
        }
        __syncthreads();
    }

    // residual = ls - recon (reuse rec arrays)
    for (int i = threadIdx.x; i < TILE; i += 256) {
        rec_re[i] = ls_re[i + HALO] - rec_re[i];
        rec_im[i] = ls_im[i + HALO] - rec_im[i];
    }
    __syncthreads();

    // ---- pass 2: h_wr -> block-MMSE (WMMA) -> * conj(ph_T) -> store ----
    const int lane = threadIdx.x & 31;
    const int wv   = threadIdx.x >> 5;
    const int col  = lane & 15;
    const int half = lane >> 4;

    for (int u = 0; u < NPORT; ++u) {
        const unsigned m_mod = (unsigned)wsi[WSI_M + u];
        const unsigned T_mod = (unsigned)wsi[WSI_T + u];
        float sts, stc; __sincosf((float)m_mod * TWO_PI_OVER_L, &sts, &stc);
        float Tss, Tsc; __sincosf((float)T_mod * TWO_PI_OVER_L, &Tss, &Tsc);

        port_avg(ls_re, ls_im, avg_re, avg_im, base, m_mod, stc, sts);
        __syncthreads();

        const int nl0 = threadIdx.x * 6;
        float s, c; __sincosf(phase_of(m_mod, (unsigned)(base + nl0)), &s, &c);
#pragma unroll
        for (int e = 0; e < 6; ++e) {
            const int nl = nl0 + e;
            float hr, hi;
            interp_at(avg_re, avg_im, base + nl, base4, &hr, &hi);
            const float rr = rec_re[nl], ri = rec_im[nl];
            wr_re[nl] = hr + rr * c - ri * s;           // h_interp + residual * ph_m
            wr_im[nl] = hi + rr * s + ri * c;
            const float nc = c * stc - s * sts;
            s = c * sts + s * stc;
            c = nc;
        }
        __syncthreads();

        // block MMSE: D(16x16) = M_pad(16x12) @ Hb(12x16) via 3 chained K=4 WMMAs,
        // separately for real and imaginary parts. wave wv owns blocks [wv*16, wv*16+16).
        v8f dre = (v8f)0.0f;
        v8f dim_ = (v8f)0.0f;
#pragma unroll
        for (int kk = 0; kk < 3; ++kk) {
            const int c0 = 4 * kk + 2 * half;           // K columns held by this half
            v2f a;                                      // A = coef * I (rows = lane%16)
            a.x = (col == c0)     ? coef : 0.0f;
            a.y = (col == c0 + 1) ? coef : 0.0f;
            const int nb = wv * 192 + col * 12 + c0;    // B rows = j index within block
            v2f br, bi;
            br.x = wr_re[nb]; br.y = wr_re[nb + 1];
            bi.x = wr_im[nb]; bi.y = wr_im[nb + 1];
            dre  = __builtin_amdgcn_wmma_f32_16x16x4_f32(false, a, false, br,
                                                         (short)0, dre,  false, false);
            dim_ = __builtin_amdgcn_wmma_f32_16x16x4_f32(false, a, false, bi,
                                                         (short)0, dim_, false, false);
        }

        // epilogue: this lane's 8 outputs are n-contiguous (row i = v + 8*half).
        // (u*L + n) is even -> pack adjacent outputs into 16B-aligned float4 stores.
        const int i0 = 8 * half;
        const int nloc0 = wv * 192 + col * 12 + i0;
        float ps, pc; __sincosf(phase_of(T_mod, (unsigned)(base + nloc0)), &ps, &pc);
        float4* outp4 = (float4*)out;
#pragma unroll
        for (int v = 0; v < 8; v += 2) {
            const int irow = i0 + v;
            // element v
            const float hr0 = dre[v],     hi0 = dim_[v];
            const float or0 = hr0 * pc + hi0 * ps;      // h_mmse * conj(ph_T)
            const float oi0 = hi0 * pc - hr0 * ps;
            float npc = pc * Tsc - ps * Tss;            // advance phasor by one sample
            ps = pc * Tss + ps * Tsc; pc = npc;
            // element v+1
            const float hr1 = dre[v + 1], hi1 = dim_[v + 1];
            const float or1 = hr1 * pc + hi1 * ps;
            const float oi1 = hi1 * pc - hr1 * ps;
            npc = pc * Tsc - ps * Tss;
            ps = pc * Tss + ps * Tsc; pc = npc;
            if (irow + 1 < 12) {                        // whole pair valid (rows 12..15 pad)
                const int n = base + wv * 192 + col * 12 + irow;
                outp4[((size_t)u * LSEQ + n) >> 1] = make_float4(or0, oi0, or1, oi1);
            }
        }
        __syncthreads();
    }
}

// ---------------------------------------------------------------- launcher
extern "C" void kernel_launch(void* const* d_in, const int* in_sizes, int n_in,
                              void* d_out, int out_size, void* d_ws, size_t ws_size,
                              hipStream_t stream) {
    (void)in_sizes; (void)n_in; (void)out_size; (void)ws_size;
    const float* lsr    = (const float*)d_in[0];
    const float* lsi    = (const float*)d_in[1];
    const int*   shifts = (const int*)d_in[2];
    float* ws  = (float*)d_ws;
    float* out = (float*)d_out;

    k_init_ws<<<1, 256, 0, stream>>>(ws);
    k_noise<<<768, 256, 0, stream>>>(lsr, lsi, ws);
    k_win_dft<<<dim3(LSEQ / 4096, NQ), 256, 0, stream>>>(lsr, lsi, ws);
    k_select<<<1, 32, 0, stream>>>(shifts, ws);
    k_main<<<LSEQ / TILE, 256, 0, stream>>>(lsr, lsi, ws, out);
}